// TextEncoder_27676769255598
// MI455X (gfx1250) — compile-verified
//
#include <hip/hip_runtime.h>
#include <math.h>
#include <stdint.h>

typedef float v2f __attribute__((ext_vector_type(2)));
typedef float v8f __attribute__((ext_vector_type(8)));
typedef int   v4i __attribute__((vector_size(16)));   // 4 x i32, as in builtin sig

// Problem constants: L=256, B=32, VOCAB=30000, K=620, D=2400
constexpr int Lc  = 256;
constexpr int Bc  = 32;
constexpr int Dc  = 2400;
constexpr int KE  = 620;           // embedding dim
constexpr int KEP = 624;           // padded to multiple of 16
constexpr int LB  = Lc * Bc;       // 8192 = M for every GEMM

// ---------- address-space helpers for async LDS copies ----------
typedef __attribute__((address_space(1))) v4i gv4i;   // global 4xi32
typedef __attribute__((address_space(3))) v4i lv4i;   // LDS 4xi32
#define AS1(p) ((gv4i*)(uintptr_t)(p))
#define AS3(p) ((lv4i*)(uint32_t)(uintptr_t)(p))

#if __has_builtin(__builtin_amdgcn_global_load_async_to_lds_b128)
#define HAVE_ASYNC_LDS 1
#else
#define HAVE_ASYNC_LDS 0
#endif

#if __has_builtin(__builtin_amdgcn_s_wait_asynccnt)
#define WAIT_ASYNC(n) __builtin_amdgcn_s_wait_asynccnt(n)
#elif HAVE_ASYNC_LDS
#define WAIT_ASYNC(n) asm volatile("s_wait_asynccnt %0" ::"i"(n) : "memory")
#else
#define WAIT_ASYNC(n) ((void)0)
#endif

// ---------------- Gather into K-padded activations ----------------
// x[m][k] (row stride KEP): emb[tokens[m]][k] for k<620, else 0
__global__ __launch_bounds__(256) void gather_emb_kernel(
    const int* __restrict__ tok, const float* __restrict__ emb,
    float* __restrict__ x) {
  int i = blockIdx.x * 256 + threadIdx.x;
  const int total = LB * KEP;
  if (i >= total) return;
  int m = i / KEP;
  int k = i - m * KEP;
  x[i] = (k < KE) ? emb[(size_t)tok[m] * KE + k] : 0.f;
}

// ---------------- Zero-pad W0 (620x9600) -> W0p (624x9600) ----------------
__global__ __launch_bounds__(256) void pad_w0_kernel(
    const float* __restrict__ W0, float* __restrict__ W0p) {
  int i = blockIdx.x * 256 + threadIdx.x;
  const int total = KEP * 9600;
  if (i >= total) return;
  int k = i / 9600;
  W0p[i] = (k < KE) ? W0[i] : 0.f;
}

// ---------------- fp32 WMMA GEMM: C[M,N] = A[M,K] * B[K,N] ----------------
// Requirements: M % 64 == 0, K % 16 == 0, N % 4 == 0 (N edge handled by
// clamped B loads + guarded C stores). Double-buffered LDS; copies done with
// GLOBAL_LOAD_ASYNC_TO_LDS_B128 when available (ASYNCcnt pipeline).
#define BM 64
#define BN 128
#define BKC 16

__global__ __launch_bounds__(256) void gemm_wmma_f32(
    const float* __restrict__ A, const float* __restrict__ Bmat,
    float* __restrict__ C, int M, int N, int K) {
  __shared__ float As[2][BM][BKC];   //  8 KB
  __shared__ float Bs[2][BKC][BN];   // 16 KB

  const int tid   = threadIdx.x;
  const int lane  = tid & 31;
  const int wave  = tid >> 5;     // 0..7
  const int waveM = wave >> 2;    // 0..1 -> 32-row strip
  const int waveN = wave & 3;     // 0..3 -> 32-col strip

  const int blockM0 = blockIdx.y * BM;
  const int blockN0 = blockIdx.x * BN;

  const int laneM = lane & 15;            // row/col within a 16x16 tile
  const int laneK = (lane >> 4) << 1;     // 0 (lanes 0-15) or 2 (lanes 16-31)

  v8f acc[2][2];
  const v8f vzero = {0.f, 0.f, 0.f, 0.f, 0.f, 0.f, 0.f, 0.f};
#pragma unroll
  for (int i = 0; i < 2; ++i)
#pragma unroll
    for (int j = 0; j < 2; ++j) acc[i][j] = vzero;

  // Per-chunk copy mapping (no divergence -> EXEC all-1s everywhere):
  //   A tile 64x16 = 256 float4 -> 1 per thread
  //   B tile 16x128 = 512 float4 -> 2 per thread (cols clamped at N edge)
  const int aRow  = tid >> 2;             // 0..63
  const int aCol4 = (tid & 3) << 2;       // 0,4,8,12
  const int bRow0 = tid >> 5;             // 0..7
  const int bRow1 = bRow0 + 8;            // 8..15
  int gnB = blockN0 + ((tid & 31) << 2);  // column of this thread's B float4
  if (gnB > N - 4) gnB = N - 4;           // clamp: junk cols are never stored
  const int bCol4 = (tid & 31) << 2;

  const float* gA = &A[(size_t)(blockM0 + aRow) * K + aCol4];

  auto issue_copy = [&](int bi, int k0) {
    const float* ga  = gA + k0;
    const float* gb0 = &Bmat[(size_t)(k0 + bRow0) * N + gnB];
    const float* gb1 = &Bmat[(size_t)(k0 + bRow1) * N + gnB];
    float* la  = &As[bi][aRow][aCol4];
    float* lb0 = &Bs[bi][bRow0][bCol4];
    float* lb1 = &Bs[bi][bRow1][bCol4];
#if HAVE_ASYNC_LDS
    __builtin_amdgcn_global_load_async_to_lds_b128(AS1(ga),  AS3(la),  0, 0);
    __builtin_amdgcn_global_load_async_to_lds_b128(AS1(gb0), AS3(lb0), 0, 0);
    __builtin_amdgcn_global_load_async_to_lds_b128(AS1(gb1), AS3(lb1), 0, 0);
#else
    *(float4*)la  = *(const float4*)ga;
    *(float4*)lb0 = *(const float4*)gb0;
    *(float4*)lb1 = *(const float4*)gb1;
#endif
  };

  const int nch = K / BKC;
  issue_copy(0, 0);

  for (int i = 0; i < nch; ++i) {
    const int cur = i & 1;
    if (i + 1 < nch) {
      issue_copy(cur ^ 1, (i + 1) * BKC);
      WAIT_ASYNC(3);   // newest 3 (next chunk) may stay in flight
    } else {
      WAIT_ASYNC(0);
    }
    __syncthreads();   // all waves' chunk-i data resident in LDS

    const float(*Asl)[BKC] = As[cur];
    const float(*Bsl)[BN]  = Bs[cur];
#pragma unroll
    for (int ks = 0; ks < 4; ++ks) {
      const int kk = ks * 4 + laneK;
      v2f afrag[2];
#pragma unroll
      for (int tm = 0; tm < 2; ++tm) {
        int row = waveM * 32 + tm * 16 + laneM;
        afrag[tm] = *(const v2f*)&Asl[row][kk];
      }
#pragma unroll
      for (int tn = 0; tn < 2; ++tn) {
        int col = waveN * 32 + tn * 16 + laneM;
        v2f bfrag;
        bfrag.x = Bsl[kk][col];
        bfrag.y = Bsl[kk + 1][col];
#pragma unroll
        for (int tm = 0; tm < 2; ++tm) {
          acc[tm][tn] = __builtin_amdgcn_wmma_f32_16x16x4_f32(
              false, afrag[tm], false, bfrag, (short)0, acc[tm][tn],
              false, false);
        }
      }
    }
    __syncthreads();   // everyone done reading buf[cur] before it is refilled
  }

  // C/D layout: VGPR r -> M = r (lanes 0-15) or r+8 (lanes 16-31), N = laneM
  const int hi = (lane >> 4) << 3;
#pragma unroll
  for (int tm = 0; tm < 2; ++tm) {
#pragma unroll
    for (int tn = 0; tn < 2; ++tn) {
      int col = blockN0 + waveN * 32 + tn * 16 + laneM;
      if (col < N) {
        int r0 = blockM0 + waveM * 32 + tm * 16 + hi;
#pragma unroll
        for (int r = 0; r < 8; ++r)
          C[(size_t)(r0 + r) * N + col] = acc[tm][tn][r];
      }
    }
  }
}

// ---------------- SRU elementwise + recurrence scan ----------------
// One thread per (b,d). U layout: U[(l*B+b)*kdim*D + j*D + d].
__global__ __launch_bounds__(256) void sru_scan_kernel(
    const float* __restrict__ U, const float* __restrict__ bias,
    const float* __restrict__ xin, float* __restrict__ h, int kdim) {
  int t = blockIdx.x * 256 + threadIdx.x;
  if (t >= Bc * Dc) return;
  int d = t % Dc;
  int b = t / Dc;
  const float bf = bias[d];
  const float br = bias[Dc + d];
  const int rowU = kdim * Dc;
  float c = 0.f;
  for (int l = 0; l < Lc; ++l) {
    size_t row = (size_t)(l * Bc + b);
    const float* u = U + row * rowU;
    float z    = u[d];
    float fg   = 1.f / (1.f + __expf(-(u[Dc + d] + bf)));
    float rg   = 1.f / (1.f + __expf(-(u[2 * Dc + d] + br)));
    float skip = (kdim == 4) ? u[3 * Dc + d] : xin[row * Dc + d];
    c = fg * c + (1.f - fg) * z;
    h[row * Dc + d] = rg * tanhf(c) + (1.f - rg) * skip;
  }
}

// ---------------- Row-wise L2 normalization ----------------
__global__ __launch_bounds__(256) void l2norm_kernel(
    const float* __restrict__ h, float* __restrict__ out) {
  __shared__ float red[256];
  const int m = blockIdx.x;
  const float* row = h + (size_t)m * Dc;
  float s = 0.f;
  for (int d = threadIdx.x; d < Dc; d += 256) {
    float v = row[d];
    s += v * v;
  }
  red[threadIdx.x] = s;
  __syncthreads();
  for (int o = 128; o > 0; o >>= 1) {
    if (threadIdx.x < o) red[threadIdx.x] += red[threadIdx.x + o];
    __syncthreads();
  }
  const float inv = rsqrtf(red[0]);
  for (int d = threadIdx.x; d < Dc; d += 256)
    out[(size_t)m * Dc + d] = row[d] * inv;
}

// ---------------- Host orchestration ----------------
extern "C" void kernel_launch(void* const* d_in, const int* in_sizes, int n_in,
                              void* d_out, int out_size, void* d_ws,
                              size_t ws_size, hipStream_t stream) {
  (void)in_sizes; (void)n_in; (void)out_size; (void)ws_size;

  const int*   tokens = (const int*)d_in[0];
  const float* emb    = (const float*)d_in[1];
  const float* W0     = (const float*)d_in[2];
  const float* b0     = (const float*)d_in[3];
  const float* W1     = (const float*)d_in[4];
  const float* b1     = (const float*)d_in[5];
  const float* W2     = (const float*)d_in[6];
  const float* b2     = (const float*)d_in[7];
  const float* W3     = (const float*)d_in[8];
  const float* b3     = (const float*)d_in[9];

  // Workspace layout (offsets multiples of 256B):
  //   X0p: 8192*624  f32 =  20,447,232 B
  //   W0p:  624*9600 f32 =  23,961,600 B
  //   U  : 8192*9600 f32 = 314,572,800 B
  //   H1 : 8192*2400 f32 =  78,643,200 B
  //   H2 : 8192*2400 f32 =  78,643,200 B     (total ~516 MB)
  char* ws = (char*)d_ws;
  float* X0p = (float*)(ws);
  float* W0p = (float*)(ws + (size_t)20447232);
  float* U   = (float*)(ws + (size_t)20447232 + 23961600);
  float* H1  = (float*)(ws + (size_t)20447232 + 23961600 + 314572800);
  float* H2  = (float*)(ws + (size_t)20447232 + 23961600 + 314572800 +
                        78643200);

  const dim3 blk(256);
  const int scanBlocks = (Bc * Dc + 255) / 256;  // 300

  // 0) Inputs -> padded workspace copies
  {
    int total = LB * KEP;
    gather_emb_kernel<<<(total + 255) / 256, blk, 0, stream>>>(tokens, emb,
                                                               X0p);
  }
  {
    int total = KEP * 9600;
    pad_w0_kernel<<<(total + 255) / 256, blk, 0, stream>>>(W0, W0p);
  }

  // 1) Layer 0: U = X0p @ W0p   (M=8192, N=9600, K=624), scan (k=4)
  {
    dim3 grid((9600 + BN - 1) / BN, LB / BM);
    gemm_wmma_f32<<<grid, blk, 0, stream>>>(X0p, W0p, U, LB, 9600, KEP);
    sru_scan_kernel<<<scanBlocks, blk, 0, stream>>>(U, b0, nullptr, H1, 4);
  }
  // 2) Layer 1: U = H1 @ W1   (N=7200, K=2400), scan (k=3, skip = H1)
  {
    dim3 grid((7200 + BN - 1) / BN, LB / BM);
    gemm_wmma_f32<<<grid, blk, 0, stream>>>(H1, W1, U, LB, 7200, Dc);
    sru_scan_kernel<<<scanBlocks, blk, 0, stream>>>(U, b1, H1, H2, 3);
  }
  // 3) Layer 2
  {
    dim3 grid((7200 + BN - 1) / BN, LB / BM);
    gemm_wmma_f32<<<grid, blk, 0, stream>>>(H2, W2, U, LB, 7200, Dc);
    sru_scan_kernel<<<scanBlocks, blk, 0, stream>>>(U, b2, H2, H1, 3);
  }
  // 4) Layer 3
  {
    dim3 grid((7200 + BN - 1) / BN, LB / BM);
    gemm_wmma_f32<<<grid, blk, 0, stream>>>(H1, W3, U, LB, 7200, Dc);
    sru_scan_kernel<<<scanBlocks, blk, 0, stream>>>(U, b3, H1, H2, 3);
  }

  // 5) L2 normalize rows into output
  l2norm_kernel<<<LB, blk, 0, stream>>>(H2, (float*)d_out);
}